// RGAT_7473243095257
// MI455X (gfx1250) — compile-verified
//
#include <hip/hip_runtime.h>
#include <hip/hip_bf16.h>
#include <math.h>

typedef __attribute__((ext_vector_type(2))) float v2f;
typedef __attribute__((ext_vector_type(8))) float v8f;

#define WPB 4   // waves per block in transform kernel

// ---------------------------------------------------------------------------
// Kernel 1: xw[n,r,:] = x[n,:] @ W[r,:,:]  (M=16/N=64/K=64 per wave via
// V_WMMA_F32_16X16X4_F32), fused with qn[n,r]=xw·q[r], kn[n,r]=xw·k[r].
// Layout assumptions (CDNA5 ISA §7.12.2, 32-bit operands):
//   A (16x4):  lane%16 = M row, lane/16 selects K-half, elem v walks K.
//   B (4x16):  lane%16 = N col, lane/16 selects K-half, elem v walks K.
//   C/D (16x16): elem v -> M = (lane/16)*8 + v, N = lane%16.
// ---------------------------------------------------------------------------
__global__ __launch_bounds__(WPB * 32) void rgat_transform_kernel(
    const float* __restrict__ X, const float* __restrict__ W,
    const float* __restrict__ qv, const float* __restrict__ kv,
    float* __restrict__ XW, float* __restrict__ qn, float* __restrict__ kn,
    int N, int R)
{
    const int lane = threadIdx.x & 31;
    const int wave = threadIdx.x >> 5;
    const int r    = blockIdx.y;
    const int m0   = (blockIdx.x * WPB + wave) * 16;
    if (m0 >= N) return;               // wave-uniform: safe w.r.t. WMMA EXEC rule

    const int lh = lane >> 4;          // K-half selector (0/1)
    const int ln = lane & 15;          // M (for A) / N (for B,C,D) index

    const float* __restrict__ Wr = W + (size_t)r * 64 * 64;
    int rowA = m0 + ln;
    if (rowA >= N) rowA = N - 1;       // clamp tail reads; stores are guarded
    const float* __restrict__ xrow = X + (size_t)rowA * 64;

    v8f acc[4] = {};                   // 4 N-tiles of the 16x64 output strip

    #pragma unroll
    for (int k0 = 0; k0 < 64; k0 += 4) {
        const int ka = k0 + lh * 2;
        v2f a;
        a[0] = xrow[ka];
        a[1] = xrow[ka + 1];
        #pragma unroll
        for (int t = 0; t < 4; ++t) {
            v2f b;
            b[0] = Wr[(size_t)ka * 64 + t * 16 + ln];
            b[1] = Wr[(size_t)(ka + 1) * 64 + t * 16 + ln];
            acc[t] = __builtin_amdgcn_wmma_f32_16x16x4_f32(
                false, a, false, b, (short)0, acc[t], false, false);
        }
    }

    // Store xw tile + accumulate per-node attention dots.
    float qa[8] = {0,0,0,0,0,0,0,0};
    float kb[8] = {0,0,0,0,0,0,0,0};
    #pragma unroll
    for (int t = 0; t < 4; ++t) {
        const float qw = qv[(size_t)r * 64 + t * 16 + ln];
        const float kw = kv[(size_t)r * 64 + t * 16 + ln];
        #pragma unroll
        for (int v = 0; v < 8; ++v) {
            const float val = acc[t][v];
            const int m = m0 + lh * 8 + v;
            if (m < N)
                XW[(((size_t)m) * R + r) * 64 + t * 16 + ln] = val;
            qa[v] += val * qw;
            kb[v] += val * kw;
        }
    }
    // Butterfly-reduce across the 16 lanes of each half (wave32 shuffles).
    #pragma unroll
    for (int off = 1; off < 16; off <<= 1) {
        #pragma unroll
        for (int v = 0; v < 8; ++v) {
            qa[v] += __shfl_xor(qa[v], off, 32);
            kb[v] += __shfl_xor(kb[v], off, 32);
        }
    }
    if (ln == 0) {
        #pragma unroll
        for (int v = 0; v < 8; ++v) {
            const int m = m0 + lh * 8 + v;
            if (m < N) {
                qn[(size_t)m * R + r] = qa[v];
                kn[(size_t)m * R + r] = kb[v];
            }
        }
    }
}

// ---------------------------------------------------------------------------
// Monotone float <-> u32 mapping for atomicMax-based segment max.
// ---------------------------------------------------------------------------
__device__ __forceinline__ unsigned enc_f32(float f) {
    unsigned u = __float_as_uint(f);
    return (u & 0x80000000u) ? ~u : (u | 0x80000000u);
}
__device__ __forceinline__ float dec_f32(unsigned u) {
    unsigned b = (u & 0x80000000u) ? (u & 0x7FFFFFFFu) : ~u;
    return __uint_as_float(b);
}

__global__ void rgat_init_kernel(float* __restrict__ accum,
                                 unsigned* __restrict__ menc,
                                 float* __restrict__ ssum, int N)
{
    const int i = blockIdx.x * blockDim.x + threadIdx.x;
    if (i < N * 64) accum[i] = 0.0f;
    if (i < N) { menc[i] = 0x007FFFFFu /* enc(-inf) */; ssum[i] = 0.0f; }
}

__global__ void rgat_edge_alpha_kernel(
    const int* __restrict__ src, const int* __restrict__ dst,
    const int* __restrict__ et, const float* __restrict__ qn,
    const float* __restrict__ kn, float* __restrict__ ebuf,
    unsigned* __restrict__ menc, int E, int R)
{
    const int e = blockIdx.x * blockDim.x + threadIdx.x;
    if (e >= E) return;
    const int s = src[e], d = dst[e], r = et[e];
    float a = qn[(size_t)d * R + r] + kn[(size_t)s * R + r];
    a = (a > 0.0f) ? a : 0.2f * a;            // leaky_relu(0.2)
    ebuf[e] = a;
    atomicMax(menc + d, enc_f32(a));
}

__global__ void rgat_edge_exp_kernel(
    const int* __restrict__ dst, float* __restrict__ ebuf,
    const unsigned* __restrict__ menc, float* __restrict__ ssum, int E)
{
    const int e = blockIdx.x * blockDim.x + threadIdx.x;
    if (e >= E) return;
    const int d = dst[e];
    float m = dec_f32(menc[d]);
    if (m < -1e30f) m = 0.0f;                 // mirrors where(isfinite(m), m, 0)
    const float ev = expf(ebuf[e] - m);
    ebuf[e] = ev;
    atomicAdd(ssum + d, ev);
}

// One wave per edge: 64-wide weighted scatter-add, 2 components per lane.
__global__ void rgat_edge_scatter_kernel(
    const int* __restrict__ src, const int* __restrict__ dst,
    const int* __restrict__ et, const float* __restrict__ ebuf,
    const float* __restrict__ ssum, const float* __restrict__ xw,
    float* __restrict__ out, int E, int R)
{
    const long long tid = (long long)blockIdx.x * blockDim.x + threadIdx.x;
    const int e = (int)(tid >> 5);
    const int lane = (int)(tid & 31);
    if (e >= E) return;
    const int s = src[e], d = dst[e], r = et[e];
    const float att = ebuf[e] / (ssum[d] + 1e-16f);
    const float* __restrict__ xs = xw + ((size_t)s * R + r) * 64;
    float* __restrict__ o = out + (size_t)d * 64;
    atomicAdd(o + lane,      att * xs[lane]);
    atomicAdd(o + lane + 32, att * xs[lane + 32]);
}

__global__ void rgat_bias_relu_kernel(const float* __restrict__ accum,
                                      const float* __restrict__ b,
                                      float* __restrict__ h, int total)
{
    const int i = blockIdx.x * blockDim.x + threadIdx.x;
    if (i >= total) return;
    const float v = accum[i] + b[i & 63];
    h[i] = v > 0.0f ? v : 0.0f;
}

// Final linear [64 -> 40] + log_softmax, one thread per node (registers only).
__global__ __launch_bounds__(128) void rgat_final_kernel(
    const float* __restrict__ h, const float* __restrict__ lw,
    const float* __restrict__ lb, float* __restrict__ out, int N)
{
    const int n = blockIdx.x * blockDim.x + threadIdx.x;
    if (n >= N) return;
    float hr[64];
    #pragma unroll
    for (int f = 0; f < 64; ++f) hr[f] = h[(size_t)n * 64 + f];
    float lg[40];
    #pragma unroll
    for (int c = 0; c < 40; ++c) {
        float a = lb[c];
        #pragma unroll
        for (int f = 0; f < 64; ++f) a += hr[f] * lw[f * 40 + c];
        lg[c] = a;
    }
    float mx = lg[0];
    #pragma unroll
    for (int c = 1; c < 40; ++c) mx = fmaxf(mx, lg[c]);
    float sum = 0.0f;
    #pragma unroll
    for (int c = 0; c < 40; ++c) sum += expf(lg[c] - mx);
    const float lse = mx + logf(sum);
    #pragma unroll
    for (int c = 0; c < 40; ++c) out[(size_t)n * 40 + c] = lg[c] - lse;
}

// ---------------------------------------------------------------------------
extern "C" void kernel_launch(void* const* d_in, const int* in_sizes, int n_in,
                              void* d_out, int out_size, void* d_ws, size_t ws_size,
                              hipStream_t stream)
{
    (void)n_in; (void)out_size; (void)ws_size;
    const float* x   = (const float*)d_in[0];
    const int*  eidx = (const int*)d_in[1];
    const int*  etyp = (const int*)d_in[2];
    const float* w1  = (const float*)d_in[3];
    const float* q1  = (const float*)d_in[4];
    const float* k1  = (const float*)d_in[5];
    const float* b1  = (const float*)d_in[6];
    const float* w2  = (const float*)d_in[7];
    const float* q2  = (const float*)d_in[8];
    const float* k2  = (const float*)d_in[9];
    const float* b2  = (const float*)d_in[10];
    const float* lw  = (const float*)d_in[11];
    const float* lb  = (const float*)d_in[12];

    const int N = in_sizes[0] / 64;
    const int E = in_sizes[2];
    const int R = in_sizes[4] / 64;   // q1 is [R,64,1]
    const int* src = eidx;
    const int* dst = eidx + E;

    // Workspace carve-out (256B aligned). xw (~102MB) is the big one and
    // is sized to live in the 192MB L2 across the edge passes.
    char* wsp = (char*)d_ws;
    auto carve = [&](size_t bytes) {
        char* p = wsp; wsp += (bytes + 255) & ~(size_t)255; return p;
    };
    float*    xw    = (float*)carve((size_t)N * R * 64 * sizeof(float));
    float*    qn    = (float*)carve((size_t)N * R * sizeof(float));
    float*    kn    = (float*)carve((size_t)N * R * sizeof(float));
    unsigned* menc  = (unsigned*)carve((size_t)N * sizeof(unsigned));
    float*    ssum  = (float*)carve((size_t)N * sizeof(float));
    float*    accum = (float*)carve((size_t)N * 64 * sizeof(float));
    float*    h1    = (float*)carve((size_t)N * 64 * sizeof(float));
    float*    h2    = (float*)carve((size_t)N * 64 * sizeof(float));
    float*    ebuf  = (float*)carve((size_t)E * sizeof(float));

    const int mtiles = (N + 15) / 16;
    const dim3 tgrid((mtiles + WPB - 1) / WPB, R);
    const int n64 = N * 64;
    const int gN64 = (n64 + 255) / 256;
    const int gE   = (E + 255) / 256;
    const int gE32 = (int)(((long long)E * 32 + 255) / 256);

    // ---- Layer 1 ----
    rgat_transform_kernel<<<tgrid, WPB * 32, 0, stream>>>(x, w1, q1, k1, xw, qn, kn, N, R);
    rgat_init_kernel<<<gN64, 256, 0, stream>>>(accum, menc, ssum, N);
    rgat_edge_alpha_kernel<<<gE, 256, 0, stream>>>(src, dst, etyp, qn, kn, ebuf, menc, E, R);
    rgat_edge_exp_kernel<<<gE, 256, 0, stream>>>(dst, ebuf, menc, ssum, E);
    rgat_edge_scatter_kernel<<<gE32, 256, 0, stream>>>(src, dst, etyp, ebuf, ssum, xw, accum, E, R);
    rgat_bias_relu_kernel<<<gN64, 256, 0, stream>>>(accum, b1, h1, n64);

    // ---- Layer 2 ----
    rgat_transform_kernel<<<tgrid, WPB * 32, 0, stream>>>(h1, w2, q2, k2, xw, qn, kn, N, R);
    rgat_init_kernel<<<gN64, 256, 0, stream>>>(accum, menc, ssum, N);
    rgat_edge_alpha_kernel<<<gE, 256, 0, stream>>>(src, dst, etyp, qn, kn, ebuf, menc, E, R);
    rgat_edge_exp_kernel<<<gE, 256, 0, stream>>>(dst, ebuf, menc, ssum, E);
    rgat_edge_scatter_kernel<<<gE32, 256, 0, stream>>>(src, dst, etyp, ebuf, ssum, xw, accum, E, R);
    rgat_bias_relu_kernel<<<gN64, 256, 0, stream>>>(accum, b2, h2, n64);

    // ---- Head ----
    rgat_final_kernel<<<(N + 127) / 128, 128, 0, stream>>>(h2, lw, lb, (float*)d_out, N);
}